// ShapeDTWLoss_31353261261221
// MI455X (gfx1250) — compile-verified
//
#include <hip/hip_runtime.h>
#include <hip/hip_bf16.h>

typedef __attribute__((ext_vector_type(2))) float v2f;
typedef __attribute__((ext_vector_type(8))) float v8f;

#define INF_ 1e10f
#define GAMMA_ 0.1f
#define LSEQ 512
#define DDIM 64
#define NDIAG (2 * LSEQ - 1)   // 1023 anti-diagonals of the 512x512 cost matrix

// ---------------------------------------------------------------------------
// softmin3 with gamma=0.1 using fast transcendentals (v_exp_f32 / v_log_f32)
// ---------------------------------------------------------------------------
__device__ __forceinline__ float softmin3(float a, float b, float c) {
    float m = fminf(fminf(a, b), c);
    float s = __expf((m - a) * 10.0f) + __expf((m - b) * 10.0f) + __expf((m - c) * 10.0f);
    return m - GAMMA_ * __logf(s);
}

// ---------------------------------------------------------------------------
// Kernel 1: cost matrix via V_WMMA_F32_16X16X4_F32, written DIAGONAL-MAJOR:
//   Cd[b][(gm+gn)*512 + gm] = ||x_gm||^2 + ||y_gn||^2 - 2 * x_gm . y_gn
// so the DP kernel's per-diagonal reads are fully coalesced (2 KB lines).
// One wave computes one 16x16 tile; block = 256 threads (8 waves);
// grid = (4, 32, nb) -> 32x32 tiles per batch.
//
// f32 WMMA A-fragment (16x4): lane l holds A[l&15][kb], A[l&15][kb+1],
// kb = (l>>4)*2.  B (4x16, col n = l&15): B[kb][n] = y[n][kb] -> contiguous
// row loads of y.  D layout: VGPR r holds M = r + 8*(l>>4), N = l&15.
// ---------------------------------------------------------------------------
__global__ __launch_bounds__(256) void sdtw_cost_wmma(
        const float* __restrict__ X,   // (nb, 512, 64), chunk base
        const float* __restrict__ Y,   // (nb, 512, 64), chunk base
        float* __restrict__ Cd) {      // (nb, 1023, 512) diagonal-major
    const int lane = threadIdx.x & 31;
    const int wave = threadIdx.x >> 5;
    const int tn = blockIdx.x * 8 + wave;   // tile col 0..31
    const int tm = blockIdx.y;              // tile row 0..31
    const int b  = blockIdx.z;

    const int hv  = lane >> 4;              // half of wave: 0 or 1
    const int idx = lane & 15;              // row of A / col of B for this lane
    const int kb  = hv * 2;

    const float* xrow = X + ((size_t)b * LSEQ + tm * 16 + idx) * DDIM;
    const float* yrow = Y + ((size_t)b * LSEQ + tn * 16 + idx) * DDIM;

    v8f acc = {};
    float sx = 0.0f, sy = 0.0f;
#pragma unroll
    for (int k0 = 0; k0 < DDIM; k0 += 4) {
        v2f a, bv;
        a.x  = xrow[k0 + kb];
        a.y  = xrow[k0 + kb + 1];
        bv.x = yrow[k0 + kb];
        bv.y = yrow[k0 + kb + 1];
        sx += a.x * a.x + a.y * a.y;
        sy += bv.x * bv.x + bv.y * bv.y;
        acc = __builtin_amdgcn_wmma_f32_16x16x4_f32(
            /*neg_a=*/false, a, /*neg_b=*/false, bv,
            /*c_mod=*/(short)0, acc, /*reuse_a=*/false, /*reuse_b=*/false);
    }
    // Each lane saw k = kb, kb+1 (mod 4); combine wave halves -> full norms.
    sx += __shfl_xor(sx, 16, 32);   // ||x_row(idx)||^2 in lanes idx, idx+16
    sy += __shfl_xor(sy, 16, 32);   // ||y_col(idx)||^2
    const float y2 = sy;            // this lane's D column is n = idx

    float* cb = Cd + (size_t)b * NDIAG * LSEQ;
#pragma unroll
    for (int r = 0; r < 8; ++r) {
        const int m = hv * 8 + r;                        // tile row for acc[r]
        const float x2 = __shfl(sx, m, 32);              // norm of x row m
        const float cval = x2 + y2 - 2.0f * acc[r];
        const int gm = tm * 16 + m;
        const int gn = tn * 16 + idx;
        cb[(size_t)(gm + gn) * LSEQ + gm] = cval;        // diagonal-major scatter
    }
}

// ---------------------------------------------------------------------------
// Kernel 2: soft-DTW wavefront DP.  One 512-thread block per batch; 3
// rotating 513-float diagonals in LDS.  Cost reads are coalesced (diagonal-
// major) and software-pipelined one diagonal ahead so the global-load
// latency overlaps the barrier + softmin chain.
// ---------------------------------------------------------------------------
__global__ __launch_bounds__(512) void sdtw_dp(
        const float* __restrict__ Cd,    // (nb, 1023, 512) chunk base
        float* __restrict__ losses) {    // chunk-offset loss array
    const int b = blockIdx.x;
    const float* C = Cd + (size_t)b * NDIAG * LSEQ;

    __shared__ float buf0[LSEQ + 1], buf1[LSEQ + 1], buf2[LSEQ + 1];
    const int tid = threadIdx.x;
    const int i = tid + 1;               // row index 1..512

    // d=0 diagonal: R[0,0]=0, rest INF.  d=1 diagonal: all INF.
    buf0[tid] = (tid == 0) ? 0.0f : INF_;
    buf1[tid] = INF_;
    if (tid == 0) { buf0[LSEQ] = INF_; buf1[LSEQ] = INF_; }
    __syncthreads();

    float* p2 = buf0;
    float* p1 = buf1;
    float* nw = buf2;

    // Pipelined cost for diagonal d: slot (i-1) of diag (d-2). Slots outside
    // the valid band hold garbage but are gated by `valid` below.
    float ccur = C[tid];                 // diagonal d=2  (dd = 0)

#pragma unroll 1
    for (int d = 2; d <= 2 * LSEQ; ++d) {
        float cnext = 0.0f;
        if (d < 2 * LSEQ) cnext = C[(size_t)(d - 1) * LSEQ + tid];  // diag d+1

        const int j = d - i;
        float v = INF_;
        if (j >= 1 && j <= LSEQ) {
            v = ccur + softmin3(p2[i - 1], p1[i - 1], p1[i]);
        }
        nw[i] = v;
        if (tid == 0) nw[0] = INF_;
        if (d == 2 * LSEQ && tid == LSEQ - 1) losses[b] = v;   // R[M,N]
        __syncthreads();
        float* t = p2; p2 = p1; p1 = nw; nw = t;               // rotate
        ccur = cnext;
    }
}

// ---------------------------------------------------------------------------
// Fallback DP (only if workspace can't hold even one cost buffer): cost
// computed on the fly; x row held in registers, y norms staged in LDS.
// ---------------------------------------------------------------------------
__global__ __launch_bounds__(512) void sdtw_dp_fly(
        const float* __restrict__ X,     // (B, 512, 64)
        const float* __restrict__ Y,     // (B, 512, 64)
        float* __restrict__ losses) {
    const int b = blockIdx.x;
    const float* x = X + (size_t)b * LSEQ * DDIM;
    const float* y = Y + (size_t)b * LSEQ * DDIM;

    __shared__ float buf0[LSEQ + 1], buf1[LSEQ + 1], buf2[LSEQ + 1];
    __shared__ float y2s[LSEQ];
    const int tid = threadIdx.x;
    const int i = tid + 1;

    float xr[DDIM];
    float x2 = 0.0f, yy = 0.0f;
#pragma unroll
    for (int k = 0; k < DDIM; ++k) {
        xr[k] = x[(size_t)(i - 1) * DDIM + k];
        x2 += xr[k] * xr[k];
        const float yv = y[(size_t)tid * DDIM + k];
        yy += yv * yv;
    }
    y2s[tid] = yy;

    buf0[tid] = (tid == 0) ? 0.0f : INF_;
    buf1[tid] = INF_;
    if (tid == 0) { buf0[LSEQ] = INF_; buf1[LSEQ] = INF_; }
    __syncthreads();

    float* p2 = buf0;
    float* p1 = buf1;
    float* nw = buf2;

#pragma unroll 1
    for (int d = 2; d <= 2 * LSEQ; ++d) {
        const int j = d - i;
        float v = INF_;
        if (j >= 1 && j <= LSEQ) {
            const float* yr = y + (size_t)(j - 1) * DDIM;
            float dot = 0.0f;
#pragma unroll
            for (int k = 0; k < DDIM; ++k) dot += xr[k] * yr[k];
            const float c = x2 + y2s[j - 1] - 2.0f * dot;
            v = c + softmin3(p2[i - 1], p1[i - 1], p1[i]);
        }
        nw[i] = v;
        if (tid == 0) nw[0] = INF_;
        if (d == 2 * LSEQ && tid == LSEQ - 1) losses[b] = v;
        __syncthreads();
        float* t = p2; p2 = p1; p1 = nw; nw = t;
    }
}

// ---------------------------------------------------------------------------
// Kernel 3: deterministic fixed-order tree reduction -> mean loss.
// ---------------------------------------------------------------------------
__global__ __launch_bounds__(64) void sdtw_reduce(
        const float* __restrict__ losses, float* __restrict__ out) {
    __shared__ float s[64];
    const int tid = threadIdx.x;
    s[tid] = losses[tid];
    __syncthreads();
#pragma unroll
    for (int off = 32; off >= 1; off >>= 1) {
        if (tid < off) s[tid] += s[tid + off];
        __syncthreads();
    }
    if (tid == 0) out[0] = s[0] * (1.0f / 64.0f);
}

// ---------------------------------------------------------------------------
// Host launcher.  Workspace layout: [0,256) = 64 per-batch losses (padded),
// [256, ...) = chunk of diagonal-major cost buffers (~2 MB per batch).
// Batches are chunked to fit ws_size; if not even one fits, use fly path.
// ---------------------------------------------------------------------------
extern "C" void kernel_launch(void* const* d_in, const int* in_sizes, int n_in,
                              void* d_out, int out_size, void* d_ws, size_t ws_size,
                              hipStream_t stream) {
    const float* pred   = (const float*)d_in[0];
    const float* target = (const float*)d_in[1];
    float* out = (float*)d_out;

    const int B = 64;
    const size_t loss_bytes = 256;
    const size_t per_batch = (size_t)NDIAG * LSEQ * sizeof(float);  // ~2 MB

    float* losses  = (float*)d_ws;
    float* costbuf = (float*)((char*)d_ws + loss_bytes);

    size_t avail = (ws_size > loss_bytes) ? (ws_size - loss_bytes) : 0;
    int chunk = (int)(avail / per_batch);
    if (chunk > B) chunk = B;

    if (chunk >= 1) {
        for (int b0 = 0; b0 < B; b0 += chunk) {
            const int nb = (B - b0 < chunk) ? (B - b0) : chunk;
            dim3 gridc(4, 32, nb);
            sdtw_cost_wmma<<<gridc, 256, 0, stream>>>(
                pred + (size_t)b0 * LSEQ * DDIM,
                target + (size_t)b0 * LSEQ * DDIM,
                costbuf);
            sdtw_dp<<<nb, LSEQ, 0, stream>>>(costbuf, losses + b0);
        }
    } else {
        sdtw_dp_fly<<<B, LSEQ, 0, stream>>>(pred, target, losses);
    }
    sdtw_reduce<<<1, 64, 0, stream>>>(losses, out);
}